// GraphAttentionLayer_76811195122279
// MI455X (gfx1250) — compile-verified
//
#include <hip/hip_runtime.h>
#include <hip/hip_bf16.h>
#include <cstdint>

#define B_    8
#define N_    1024
#define FIN_  512
#define FOUT_ 256
#define ALPHA_ 0.2f
#define EPS_   1e-5f
#define NEG_   (-9000000000000000.0f)

typedef __attribute__((ext_vector_type(2))) float v2f;
typedef __attribute__((ext_vector_type(4))) float v4f;
typedef __attribute__((ext_vector_type(8))) float v8f;

__device__ __forceinline__ float lrelu(float x) {
  return x > 0.f ? x : ALPHA_ * x;
}

// ---------------------------------------------------------------------------
// Kernel 0: WT[o][k] = W[k][o]  (one-shot 512 KB transpose so GEMM B-fragments
// become contiguous b64 loads)
// ---------------------------------------------------------------------------
__global__ void w_transpose(const float* __restrict__ W, float* __restrict__ WT) {
  const int idx = blockIdx.x * 256 + threadIdx.x;   // 0 .. FIN*FOUT-1
  const int o = idx / FIN_;
  const int k = idx % FIN_;
  WT[idx] = W[(size_t)k * FOUT_ + o];
}

// ---------------------------------------------------------------------------
// Kernel 1: WhT[b][o][n] = (h @ W)^T via V_WMMA_F32_16X16X4_F32.
// h strip staged in LDS (conflict-free ds_load_2addr_b64 A-frags);
// B-frag = single global_load_b64 from WT; result stored transposed as
// 2x global_store_b128 per lane (8 contiguous n per lane).
// ---------------------------------------------------------------------------
#define HS_STR 516   // 512 + 4 pad -> bank shift of 4 per row (conflict-free)
__global__ void wh_gemm(const float* __restrict__ h, const float* __restrict__ WT,
                        float* __restrict__ WhT) {
  __shared__ float hs[16 * HS_STR];   // 33 KB

  const int t      = threadIdx.x;
  const int lane   = t & 31;
  const int wave   = t >> 5;
  const int o_tile = blockIdx.x * 8 + wave;   // 0..15
  const int m_tile = blockIdx.y;              // 0..63
  const int b      = blockIdx.z;              // 0..7
  const int l15    = lane & 15;
  const int khalf  = (lane >> 4) * 2;         // 0 or 2

  // cooperative, coalesced fill of the 16 x FIN h strip
  {
    const int r = t >> 4;
    const float* hr = h + ((size_t)b * N_ + m_tile * 16 + r) * FIN_;
    for (int cidx = t & 15; cidx < FIN_; cidx += 16)
      hs[r * HS_STR + cidx] = hr[cidx];
  }
  __syncthreads();

  const int col = o_tile * 16 + l15;
  const float* wtrow = WT + (size_t)col * FIN_;

  v8f c = {0.f, 0.f, 0.f, 0.f, 0.f, 0.f, 0.f, 0.f};
  for (int k = 0; k < FIN_; k += 4) {
    v2f af = *(const v2f*)&hs[l15 * HS_STR + k + khalf];   // ds_load (paired)
    v2f bf = *(const v2f*)&wtrow[k + khalf];               // single b64
    c = __builtin_amdgcn_wmma_f32_16x16x4_f32(false, af, false, bf,
                                              (short)0, c, false, false);
  }

  // transposed store: lane owns rows n = m_tile*16 + rb .. rb+7 at fixed col
  const int rb = 8 * (lane >> 4);
  float* outp = WhT + ((size_t)b * FOUT_ + col) * N_ + m_tile * 16 + rb;
  v4f lo = {c[0], c[1], c[2], c[3]};
  v4f hi = {c[4], c[5], c[6], c[7]};
  *(v4f*)(outp)     = lo;   // global_store_b128
  *(v4f*)(outp + 4) = hi;   // global_store_b128
}

// ---------------------------------------------------------------------------
// Kernel 2: f1/f2 from WhT — thread per node, coalesced across n,
// av[o] is wave-uniform -> scalar loads. Deterministic sequential sum.
// ---------------------------------------------------------------------------
__global__ void attn_vec(const float* __restrict__ WhT, const float* __restrict__ av,
                         float* __restrict__ f1, float* __restrict__ f2) {
  const int idx = blockIdx.x * 256 + threadIdx.x;   // 0..B*N-1
  const int b   = idx >> 10;                        // N_ == 1024
  const int n   = idx & (N_ - 1);
  const float* base = WhT + (size_t)b * FOUT_ * N_ + n;
  float s1 = 0.f, s2 = 0.f;
  for (int o = 0; o < FOUT_; ++o) {
    float v = base[(size_t)o * N_];
    s1 += v * av[o];
    s2 += v * av[FOUT_ + o];
  }
  f1[idx] = s1;
  f2[idx] = s2;
}

// ---------------------------------------------------------------------------
// Kernel 3 (fused): masked leaky-relu logits -> softmax -> WMMA h' = P @ Wh
// -> LayerNorm -> exact GELU.  One block per (16-row strip, batch).
// P strip (16 x N, stride 1028 -> conflict-free) in 64 KB dynamic LDS;
// f2 pulled in with global_load_async_to_lds_b128 (ASYNCcnt path);
// softmax passes vectorized to b128; B-frags are single b64 from WhT.
// ---------------------------------------------------------------------------
#define PSTR 1028
#define HSTR 257
extern __shared__ float Psmem[];

__global__ void gat_fused(const float* __restrict__ adj, const float* __restrict__ WhT,
                          const float* __restrict__ f1, const float* __restrict__ f2,
                          const float* __restrict__ gamma, const float* __restrict__ beta,
                          float* __restrict__ out) {
  __shared__ float red[512];
  __shared__ float rowmax[16];
  __shared__ float rowinv[16];
  __shared__ __align__(16) float f2s[N_];

  const int t      = threadIdx.x;       // 0..255
  const int i_tile = blockIdx.x;        // 0..63
  const int b      = blockIdx.y;        // 0..7
  const int i_base = i_tile * 16;

  const float* adjS = adj + ((size_t)b * N_ + i_base) * N_;   // 16 x N strip
  const float* f1b  = f1 + (size_t)b * N_ + i_base;
  const float* f2b  = f2 + (size_t)b * N_;

  // ---- async copy of f2 (4 KB) into LDS: ASYNCcnt path ----
  if (t < 32) {
    const unsigned ldsbase = (unsigned)(uintptr_t)(&f2s[0]);
    for (int i = t * 4; i < N_; i += 128) {
      const float* gp = f2b + i;
      unsigned lp = ldsbase + (unsigned)(i * 4);
      asm volatile("global_load_async_to_lds_b128 %0, %1, off"
                   :: "v"(lp), "v"(gp) : "memory");
    }
#if __has_builtin(__builtin_amdgcn_s_wait_asynccnt)
    __builtin_amdgcn_s_wait_asynccnt(0);
#else
    asm volatile("s_wait_asynccnt 0" ::: "memory");
#endif
  }
  __syncthreads();

  // ---- pass 1: logits e -> P (b128), row max ----
  {
    const int r = t >> 4;
    const int j0 = (t & 15) * 4;
    const float f1v = f1b[r];
    const float* adjR = adjS + (size_t)r * N_;
    float m = -INFINITY;
    for (int j = j0; j < N_; j += 64) {
      v4f a4 = *(const v4f*)&adjR[j];     // global_load_b128, coalesced
      v4f fv = *(const v4f*)&f2s[j];      // ds_load_b128
      v4f e;
#pragma unroll
      for (int q = 0; q < 4; ++q) {
        float ev = a4[q] > 0.f ? lrelu(f1v + fv[q]) : NEG_;
        e[q] = ev;
        m = fmaxf(m, ev);
      }
      *(v4f*)&Psmem[r * PSTR + j] = e;    // ds_store_b128
    }
    red[t] = m;
  }
  __syncthreads();
  if (t < 16) {
    float m = red[t * 16];
#pragma unroll
    for (int k = 1; k < 16; ++k) m = fmaxf(m, red[t * 16 + k]);
    rowmax[t] = m;
  }
  __syncthreads();

  // ---- pass 2: P <- exp(e - max) in place (b128), row sum ----
  {
    const int r = t >> 4;
    const int j0 = (t & 15) * 4;
    const float rm = rowmax[r];
    float s = 0.f;
    for (int j = j0; j < N_; j += 64) {
      v4f e = *(const v4f*)&Psmem[r * PSTR + j];
      v4f p;
#pragma unroll
      for (int q = 0; q < 4; ++q) {
        p[q] = __expf(e[q] - rm);
        s += p[q];
      }
      *(v4f*)&Psmem[r * PSTR + j] = p;
    }
    red[t] = s;
  }
  __syncthreads();
  if (t < 16) {
    float s = 0.f;
#pragma unroll
    for (int k = 0; k < 16; ++k) s += red[t * 16 + k];
    rowinv[t] = 1.0f / s;
  }
  __syncthreads();

  // ---- WMMA pass: h'[16 x 256] = P(16 x N) @ Wh(N x 256), B from WhT ----
  {
    const int lane  = t & 31;
    const int wave  = t >> 5;                 // o-tiles {wave, wave+8}
    const int l15   = lane & 15;
    const int khalf = (lane >> 4) * 2;
    const int col0  = wave * 16 + l15;
    const int col1  = col0 + 128;
    const float* pp  = &Psmem[l15 * PSTR];
    const float* wt0 = WhT + ((size_t)b * FOUT_ + col0) * N_;
    const float* wt1 = WhT + ((size_t)b * FOUT_ + col1) * N_;

    v8f c0 = {0.f, 0.f, 0.f, 0.f, 0.f, 0.f, 0.f, 0.f};
    v8f c1 = {0.f, 0.f, 0.f, 0.f, 0.f, 0.f, 0.f, 0.f};

    for (int jj = 0; jj < N_; jj += 4) {
      const int j0 = jj + khalf;
      v2f af = *(const v2f*)&pp[j0];          // conflict-free ds_load (paired)
      v2f b0 = *(const v2f*)&wt0[j0];         // single global_load_b64
      v2f b1 = *(const v2f*)&wt1[j0];         // single global_load_b64

      c0 = __builtin_amdgcn_wmma_f32_16x16x4_f32(false, af, false, b0,
                                                 (short)0, c0, false, false);
      c1 = __builtin_amdgcn_wmma_f32_16x16x4_f32(false, af, false, b1,
                                                 (short)0, c1, false, false);
    }
    __syncthreads();   // everyone done reading P; reuse buffer as h' tile

    float* htile = Psmem;
    const int rb = 8 * (lane >> 4);
#pragma unroll
    for (int v = 0; v < 8; ++v) {
      const int r = v + rb;
      const float inv = rowinv[r];
      htile[r * HSTR + col0] = c0[v] * inv;
      htile[r * HSTR + col1] = c1[v] * inv;
    }
  }
  __syncthreads();

  // ---- epilogue: LayerNorm (population var) + exact GELU ----
  const float* htile = Psmem;
  {
    const int r  = t >> 4;
    const int cs = (t & 15) * 16;
    float ps = 0.f, ps2 = 0.f;
#pragma unroll
    for (int k = 0; k < 16; ++k) {
      float x = htile[r * HSTR + cs + k];
      ps += x; ps2 += x * x;
    }
    red[t]       = ps;
    red[256 + t] = ps2;
  }
  __syncthreads();
  if (t < 16) {
    float s = 0.f, s2 = 0.f;
#pragma unroll
    for (int k = 0; k < 16; ++k) { s += red[t * 16 + k]; s2 += red[256 + t * 16 + k]; }
    float mu  = s * (1.0f / FOUT_);
    float var = s2 * (1.0f / FOUT_) - mu * mu;
    rowmax[t] = mu;                       // reuse as mean
    rowinv[t] = rsqrtf(var + EPS_);       // reuse as rstd
  }
  __syncthreads();
  {
    const int r  = t >> 4;
    const int cs = (t & 15) * 16;
    const float mu = rowmax[r], rstd = rowinv[r];
    float* o = out + ((size_t)b * N_ + i_base + r) * FOUT_;
#pragma unroll
    for (int k = 0; k < 16; ++k) {
      const int cidx = cs + k;
      float x = htile[r * HSTR + cidx];
      float y = (x - mu) * rstd * gamma[cidx] + beta[cidx];
      o[cidx] = 0.5f * y * (1.0f + erff(y * 0.70710678118654752f));
    }
  }
}

// ---------------------------------------------------------------------------
extern "C" void kernel_launch(void* const* d_in, const int* in_sizes, int n_in,
                              void* d_out, int out_size, void* d_ws, size_t ws_size,
                              hipStream_t stream) {
  (void)in_sizes; (void)n_in; (void)out_size; (void)ws_size;
  const float* h     = (const float*)d_in[0];  // (B,N,FIN)
  const float* adj   = (const float*)d_in[1];  // (B,N,N)
  // d_in[2] = q_type, d_in[3] = pos : unused by reference
  const float* W     = (const float*)d_in[4];  // (FIN,FOUT)
  const float* avec  = (const float*)d_in[5];  // (2*FOUT,1)
  const float* gamma = (const float*)d_in[6];  // (FOUT,)
  const float* beta  = (const float*)d_in[7];  // (FOUT,)
  float* out = (float*)d_out;

  float* WhT = (float*)d_ws;                         // B*FOUT*N floats = 8 MB
  float* f1  = WhT + (size_t)B_ * FOUT_ * N_;        // B*N floats
  float* f2  = f1 + (size_t)B_ * N_;                 // B*N floats
  float* WT  = f2 + (size_t)B_ * N_;                 // FOUT*FIN floats = 512 KB

  const size_t psize = (size_t)16 * PSTR * sizeof(float);   // 65792 B dynamic LDS

  w_transpose<<<dim3(FIN_ * FOUT_ / 256),        256, 0,     stream>>>(W, WT);
  wh_gemm    <<<dim3(FOUT_ / 16 / 8, N_ / 16, B_), 256, 0,   stream>>>(h, WT, WhT);
  attn_vec   <<<dim3(B_ * N_ / 256),             256, 0,     stream>>>(WhT, avec, f1, f2);
  gat_fused  <<<dim3(N_ / 16, B_),               256, psize, stream>>>(adj, WhT, f1, f2,
                                                                       gamma, beta, out);
}